// Encoder_new_1176821039652
// MI455X (gfx1250) — compile-verified
//
#include <hip/hip_runtime.h>
#include <hip/hip_bf16.h>
#include <stdint.h>

#define N_NODES 100000
#define X_DIM   512
#define HID     64
#define N_EDGES 1000000

typedef float v2f __attribute__((ext_vector_type(2)));
typedef float v8f __attribute__((ext_vector_type(8)));

// ---------------------------------------------------------------------------
// Kernel 1: h = relu(x @ W1^T + b1)   [N,512]x[512,64] via V_WMMA_F32_16X16X4_F32
// One wave computes a 16x64 output strip (4 accumulators). A/B fragments are
// per-lane float2 loads per ISA 7.12.2 layout. x loads are non-temporal so the
// 205MB stream does not evict L2-resident h.
// ---------------------------------------------------------------------------
__global__ __launch_bounds__(256) void k_gemm1(
    const float* __restrict__ x, const float* __restrict__ W1,
    const float* __restrict__ bias1, float* __restrict__ h, int nTiles)
{
  const int wave = blockIdx.x * 8 + (threadIdx.x >> 5);
  if (wave >= nTiles) return;                 // wave-uniform: EXEC stays all-1s
  const int lane = threadIdx.x & 31;
  const int m    = lane & 15;
  const int koff = (lane >> 4) << 1;

  const float* xr = x  + (size_t)(wave * 16 + m) * X_DIM + koff;
  const float* w0 = W1 + (size_t)( 0 + m) * X_DIM + koff;
  const float* w1 = W1 + (size_t)(16 + m) * X_DIM + koff;
  const float* w2 = W1 + (size_t)(32 + m) * X_DIM + koff;
  const float* w3 = W1 + (size_t)(48 + m) * X_DIM + koff;

  v8f c0 = {}, c1 = {}, c2 = {}, c3 = {};
  #pragma unroll 8
  for (int k = 0; k < X_DIM; k += 4) {
    v2f a  = __builtin_nontemporal_load((const v2f*)(xr + k));
    v2f b0 = *(const v2f*)(w0 + k);
    v2f b1 = *(const v2f*)(w1 + k);
    v2f b2 = *(const v2f*)(w2 + k);
    v2f b3 = *(const v2f*)(w3 + k);
    c0 = __builtin_amdgcn_wmma_f32_16x16x4_f32(false, a, false, b0, (short)0, c0, false, false);
    c1 = __builtin_amdgcn_wmma_f32_16x16x4_f32(false, a, false, b1, (short)0, c1, false, false);
    c2 = __builtin_amdgcn_wmma_f32_16x16x4_f32(false, a, false, b2, (short)0, c2, false, false);
    c3 = __builtin_amdgcn_wmma_f32_16x16x4_f32(false, a, false, b3, (short)0, c3, false, false);
  }

  const int rbase = wave * 16 + ((lane >> 4) << 3);
  const float e0 = bias1[ 0 + m], e1 = bias1[16 + m];
  const float e2 = bias1[32 + m], e3 = bias1[48 + m];
  #pragma unroll
  for (int r = 0; r < 8; ++r) {
    size_t o = (size_t)(rbase + r) * HID + m;
    h[o +  0] = fmaxf(c0[r] + e0, 0.0f);
    h[o + 16] = fmaxf(c1[r] + e1, 0.0f);
    h[o + 32] = fmaxf(c2[r] + e2, 0.0f);
    h[o + 48] = fmaxf(c3[r] + e3, 0.0f);
  }
}

// ---------------------------------------------------------------------------
// CSR build: zero counts -> count (int atomics) -> scan (1 block) -> fill
// ---------------------------------------------------------------------------
__global__ void k_zero_i32(int* __restrict__ p, int n) {
  int i = blockIdx.x * blockDim.x + threadIdx.x;
  if (i < n) p[i] = 0;
}

__global__ void k_count(const long long* __restrict__ ei, int* __restrict__ cnt, int E) {
  int e = blockIdx.x * blockDim.x + threadIdx.x;
  if (e >= E) return;
  atomicAdd(&cnt[(int)ei[(size_t)E + e]], 1);
}

__global__ __launch_bounds__(1024) void k_scan(
    const int* __restrict__ cnt, int* __restrict__ rowptr,
    int* __restrict__ wptr, int n)
{
  __shared__ int part[1024];
  __shared__ int base[1024];
  const int tid   = threadIdx.x;
  const int chunk = (n + 1023) / 1024;
  const int lo = tid * chunk;
  const int hi = (lo + chunk < n) ? (lo + chunk) : n;
  int s = 0;
  for (int i = lo; i < hi; ++i) s += cnt[i];
  part[tid] = s;
  __syncthreads();
  if (tid == 0) {
    int run = 0;
    for (int i = 0; i < 1024; ++i) { base[i] = run; run += part[i]; }
    rowptr[n] = run;
  }
  __syncthreads();
  int run = base[tid];
  for (int i = lo; i < hi; ++i) { rowptr[i] = run; wptr[i] = run; run += cnt[i]; }
}

__global__ void k_fill(const long long* __restrict__ ei, int* __restrict__ wptr,
                       int* __restrict__ colidx, int E)
{
  int e = blockIdx.x * blockDim.x + threadIdx.x;
  if (e >= E) return;
  int src = (int)ei[e];
  int dst = (int)ei[(size_t)E + e];
  int pos = atomicAdd(&wptr[dst], 1);
  colidx[pos] = src;
}

// ---------------------------------------------------------------------------
// Mean aggregation: one wave per node, lane owns channels {lane, lane+32}.
// Gathers are coalesced 128B reads of L2-resident h; zero float atomics.
// ---------------------------------------------------------------------------
__global__ __launch_bounds__(256) void k_aggregate(
    const float* __restrict__ h, const int* __restrict__ rowptr,
    const int* __restrict__ colidx, float* __restrict__ mean, int n)
{
  const int node = blockIdx.x * 8 + (threadIdx.x >> 5);
  if (node >= n) return;
  const int lane = threadIdx.x & 31;
  const int beg = rowptr[node], end = rowptr[node + 1];
  float a0 = 0.0f, a1 = 0.0f;
  #pragma unroll 2
  for (int j = beg; j < end; ++j) {
    const float* hs = h + (size_t)colidx[j] * HID + lane;
    a0 += hs[0];
    a1 += hs[32];
  }
  const float inv = 1.0f / fmaxf((float)(end - beg), 1.0f);
  mean[(size_t)node * HID + lane     ] = a0 * inv;
  mean[(size_t)node * HID + lane + 32] = a1 * inv;
}

// ---------------------------------------------------------------------------
// Kernel 3: out = mean @ Wl^T + bl + h @ Wr^T  (two K=64 GEMMs, shared C accum)
// ---------------------------------------------------------------------------
__global__ __launch_bounds__(256) void k_gemm2(
    const float* __restrict__ h, const float* __restrict__ mean,
    const float* __restrict__ Wl, const float* __restrict__ blv,
    const float* __restrict__ Wr, float* __restrict__ out, int nTiles)
{
  const int wave = blockIdx.x * 8 + (threadIdx.x >> 5);
  if (wave >= nTiles) return;
  const int lane = threadIdx.x & 31;
  const int m    = lane & 15;
  const int koff = (lane >> 4) << 1;

  const float* hr = h    + (size_t)(wave * 16 + m) * HID + koff;
  const float* mr = mean + (size_t)(wave * 16 + m) * HID + koff;
  const float* l0 = Wl + (size_t)( 0 + m) * HID + koff;
  const float* l1 = Wl + (size_t)(16 + m) * HID + koff;
  const float* l2 = Wl + (size_t)(32 + m) * HID + koff;
  const float* l3 = Wl + (size_t)(48 + m) * HID + koff;
  const float* r0 = Wr + (size_t)( 0 + m) * HID + koff;
  const float* r1 = Wr + (size_t)(16 + m) * HID + koff;
  const float* r2 = Wr + (size_t)(32 + m) * HID + koff;
  const float* r3 = Wr + (size_t)(48 + m) * HID + koff;

  v8f c0 = {}, c1 = {}, c2 = {}, c3 = {};
  #pragma unroll
  for (int k = 0; k < HID; k += 4) {
    v2f am = *(const v2f*)(mr + k);
    v2f ah = *(const v2f*)(hr + k);
    v2f b;
    b = *(const v2f*)(l0 + k); c0 = __builtin_amdgcn_wmma_f32_16x16x4_f32(false, am, false, b, (short)0, c0, false, false);
    b = *(const v2f*)(r0 + k); c0 = __builtin_amdgcn_wmma_f32_16x16x4_f32(false, ah, false, b, (short)0, c0, false, false);
    b = *(const v2f*)(l1 + k); c1 = __builtin_amdgcn_wmma_f32_16x16x4_f32(false, am, false, b, (short)0, c1, false, false);
    b = *(const v2f*)(r1 + k); c1 = __builtin_amdgcn_wmma_f32_16x16x4_f32(false, ah, false, b, (short)0, c1, false, false);
    b = *(const v2f*)(l2 + k); c2 = __builtin_amdgcn_wmma_f32_16x16x4_f32(false, am, false, b, (short)0, c2, false, false);
    b = *(const v2f*)(r2 + k); c2 = __builtin_amdgcn_wmma_f32_16x16x4_f32(false, ah, false, b, (short)0, c2, false, false);
    b = *(const v2f*)(l3 + k); c3 = __builtin_amdgcn_wmma_f32_16x16x4_f32(false, am, false, b, (short)0, c3, false, false);
    b = *(const v2f*)(r3 + k); c3 = __builtin_amdgcn_wmma_f32_16x16x4_f32(false, ah, false, b, (short)0, c3, false, false);
  }

  const int rbase = wave * 16 + ((lane >> 4) << 3);
  const float e0 = blv[ 0 + m], e1 = blv[16 + m];
  const float e2 = blv[32 + m], e3 = blv[48 + m];
  #pragma unroll
  for (int r = 0; r < 8; ++r) {
    size_t o = (size_t)(rbase + r) * HID + m;
    out[o +  0] = c0[r] + e0;
    out[o + 16] = c1[r] + e1;
    out[o + 32] = c2[r] + e2;
    out[o + 48] = c3[r] + e3;
  }
}

// ---------------------------------------------------------------------------
// Workspace layout (bytes, all regions >=64B aligned):
//   h      : [N*HID] f32 @ 0          (25,600,000)
//   mean   : [N*HID] f32 @ 25,600,000 (25,600,000)
//   cnt    : [N]     i32 @ 51,200,000 (  400,000)
//   rowptr : [N+1]   i32 @ 51,600,000 (  400,004)
//   wptr   : [N]     i32 @ 52,000,064 (  400,000)
//   col    : [E]     i32 @ 52,400,064 (4,000,000)   total ~56.4 MB
// ---------------------------------------------------------------------------
extern "C" void kernel_launch(void* const* d_in, const int* in_sizes, int n_in,
                              void* d_out, int out_size, void* d_ws, size_t ws_size,
                              hipStream_t stream) {
  (void)in_sizes; (void)n_in; (void)out_size; (void)ws_size;
  const float*     x   = (const float*)d_in[0];
  const long long* ei  = (const long long*)d_in[1];
  const float*     W1  = (const float*)d_in[2];
  const float*     b1  = (const float*)d_in[3];
  const float*     Wl  = (const float*)d_in[4];
  const float*     bl  = (const float*)d_in[5];
  const float*     Wr  = (const float*)d_in[6];
  float*           out = (float*)d_out;

  char* ws = (char*)d_ws;
  float* h      = (float*)(ws + 0);
  float* mean   = (float*)(ws + 25600000);
  int*   cnt    = (int*)  (ws + 51200000);
  int*   rowptr = (int*)  (ws + 51600000);
  int*   wptr   = (int*)  (ws + 52000064);
  int*   colidx = (int*)  (ws + 52400064);

  const int nTiles = N_NODES / 16;  // 6250, exact

  k_gemm1    <<<(nTiles + 7) / 8,        256,  0, stream>>>(x, W1, b1, h, nTiles);
  k_zero_i32 <<<(N_NODES + 255) / 256,   256,  0, stream>>>(cnt, N_NODES);
  k_count    <<<(N_EDGES + 255) / 256,   256,  0, stream>>>(ei, cnt, N_EDGES);
  k_scan     <<<1,                       1024, 0, stream>>>(cnt, rowptr, wptr, N_NODES);
  k_fill     <<<(N_EDGES + 255) / 256,   256,  0, stream>>>(ei, wptr, colidx, N_EDGES);
  k_aggregate<<<(N_NODES + 7) / 8,       256,  0, stream>>>(h, rowptr, colidx, mean, N_NODES);
  k_gemm2    <<<(nTiles + 7) / 8,        256,  0, stream>>>(h, mean, Wl, bl, Wr, out, nTiles);
}